// PenaltyMethod_14113262534973
// MI455X (gfx1250) — compile-verified
//
#include <hip/hip_runtime.h>

typedef float v2f __attribute__((ext_vector_type(2)));
typedef float v8f __attribute__((ext_vector_type(8)));

constexpr int B_ = 4, V_ = 8192, F_ = 16384, P_ = 4, E_ = 2048;
constexpr int ROWS_PER_WG = 128;          // 8 waves * 16 rows
constexpr int COL_TILES   = E_ / 16;      // 128
constexpr int FINF_BITS   = 0x7f800000;   // +inf; int-compare-safe for d2 >= 0

// ---------------------------------------------------------------------------
// Kernel 1: pack A (projected vertices) and B (edge points) operands for the
// distance-expansion WMMA:  d2 = A_row . B_col with
//   A_row = [px, py, px^2+py^2, 1],  B_col = [-2e0, -2e1, 1, e0^2+e1^2]
// ---------------------------------------------------------------------------
__global__ __launch_bounds__(256)
void pm_prep_kernel(const float* __restrict__ y, const float* __restrict__ pmat,
                    const float* __restrict__ em, float* __restrict__ apack,
                    float* __restrict__ bpack) {
    int idx = blockIdx.x * 256 + threadIdx.x;
    if (idx < B_ * P_ * V_) {
        int v  = idx % V_;
        int bp = idx / V_;
        int p  = bp % P_;
        int b  = bp / P_;
        const float* yv = y + ((size_t)b * V_ + v) * 3;
        float y0 = yv[0], y1 = yv[1], y2 = yv[2];
        const float* M = pmat + p * 12;
        float pr0 = M[0] * y0 + M[1] * y1 + M[2]  * y2 + M[3];
        float pr1 = M[4] * y0 + M[5] * y1 + M[6]  * y2 + M[7];
        float pr2 = M[8] * y0 + M[9] * y1 + M[10] * y2 + M[11];
        float px = pr0 / pr2;
        float py = pr1 / pr2;
        float4 a = make_float4(px, py, px * px + py * py, 1.0f);
        *(float4*)(apack + (size_t)idx * 4) = a;
    }
    if (idx < B_ * P_ * E_) {
        float e0 = em[(size_t)idx * 2 + 0];
        float e1 = em[(size_t)idx * 2 + 1];
        float4 bb = make_float4(-2.0f * e0, -2.0f * e1, 1.0f, e0 * e0 + e1 * e1);
        *(float4*)(bpack + (size_t)idx * 4) = bb;
    }
}

// ---------------------------------------------------------------------------
// Kernel 2: init global column-min buffer to +inf (stored as int bits)
// ---------------------------------------------------------------------------
__global__ __launch_bounds__(256)
void pm_init_kernel(int* __restrict__ colmin) {
    int idx = blockIdx.x * 256 + threadIdx.x;
    if (idx < B_ * P_ * E_) colmin[idx] = FINF_BITS;
}

// ---------------------------------------------------------------------------
// Kernel 3: WMMA chamfer.  One WG = (b,p, 128-row chunk); 8 waves, each wave
// owns a 16-row A tile and sweeps all 128 column tiles with
// v_wmma_f32_16x16x4_f32.  All min-reductions operate on float *bits* as
// signed ints (valid for non-negative d2): no NaN-canonicalize VALU ops and
// deterministic integer atomic-min.  B fragments are software-pipelined.
// ---------------------------------------------------------------------------
__global__ __launch_bounds__(256)
void pm_chamfer_kernel(const float* __restrict__ apack, const float* __restrict__ bpack,
                       int* __restrict__ rowmin, int* __restrict__ colminG) {
    __shared__ int colminLds[E_];           // 8 KB

    int tid = threadIdx.x;
    for (int i = tid; i < E_; i += 256) colminLds[i] = FINF_BITS;
    __syncthreads();

    const int chunks = V_ / ROWS_PER_WG;    // 64
    int wg    = blockIdx.x;
    int chunk = wg % chunks;
    int bp    = wg / chunks;                // b*P + p
    int lane  = tid & 31;
    int wave  = tid >> 5;
    int rowbase = chunk * ROWS_PER_WG + wave * 16;

    const float* Abase = apack + (size_t)bp * V_ * 4;
    const float* Bbase = bpack + (size_t)bp * E_ * 4;

    // A fragment: 16x4 f32 -> 2 VGPRs/lane.
    // lanes 0-15: row=rowbase+lane, K={0,1}; lanes 16-31: row=rowbase+lane-16, K={2,3}
    int colidx = lane & 15;
    int koff   = (lane >> 4) << 1;          // 0 or 2
    v2f afrag  = *(const v2f*)(Abase + (size_t)(rowbase + colidx) * 4 + koff);

    const float* Bk = Bbase + koff;
    v2f bnext = *(const v2f*)(Bk + (size_t)colidx * 4);

    int rowaccI[8];
#pragma unroll
    for (int i = 0; i < 8; ++i) rowaccI[i] = FINF_BITS;

#pragma unroll 4
    for (int ct = 0; ct < COL_TILES; ++ct) {
        v2f bfrag = bnext;
        int nct = (ct + 1) & (COL_TILES - 1);                 // wrap: cheap AND
        bnext = *(const v2f*)(Bk + (size_t)(nct * 16 + colidx) * 4);

        v8f c = {};
        c = __builtin_amdgcn_wmma_f32_16x16x4_f32(
                /*neg_a=*/false, afrag, /*neg_b=*/false, bfrag,
                /*c_mod=*/(short)0, c, /*reuse_a=*/false, /*reuse_b=*/false);

        int ci[8];
#pragma unroll
        for (int i = 0; i < 8; ++i) ci[i] = __float_as_int(c[i]);

        // column min within this lane half (8 rows), all 32 lanes atomically
        // combine both halves at the same 16 LDS addresses
        int cm = min(min(min(ci[0], ci[1]), min(ci[2], ci[3])),
                     min(min(ci[4], ci[5]), min(ci[6], ci[7])));
        atomicMin(&colminLds[ct * 16 + colidx], cm);

        // running row-min accumulator (register-resident)
#pragma unroll
        for (int i = 0; i < 8; ++i) rowaccI[i] = min(rowaccI[i], ci[i]);
    }

    // Row reduction: min across the 16 lanes of each half (N dimension)
#pragma unroll
    for (int m = 1; m <= 8; m <<= 1) {
#pragma unroll
        for (int i = 0; i < 8; ++i)
            rowaccI[i] = min(rowaccI[i], __shfl_xor(rowaccI[i], m, 32));
    }
    if ((lane & 15) == 0) {
        // lane 0 -> rows rowbase+0..7; lane 16 -> rows rowbase+8..15
        int r0 = rowbase + ((lane >> 4) << 3);
        int* outp = rowmin + (size_t)bp * V_ + r0;
#pragma unroll
        for (int i = 0; i < 8; ++i) outp[i] = rowaccI[i];
    }

    __syncthreads();
    int* cg = colminG + (size_t)bp * E_;
    for (int i = tid; i < E_; i += 256) atomicMin(&cg[i], colminLds[i]);
}

// ---------------------------------------------------------------------------
// Kernel 4: one block per batch: L2 term, face volumes, chamfer means, output.
// Deterministic fixed-order tree reductions in LDS.
// ---------------------------------------------------------------------------
__device__ float pm_block_reduce(float v, float* red, int tid) {
    red[tid] = v;
    __syncthreads();
    for (int s = 128; s > 0; s >>= 1) {
        if (tid < s) red[tid] += red[tid + s];
        __syncthreads();
    }
    float r = red[0];
    __syncthreads();
    return r;
}

__global__ __launch_bounds__(256)
void pm_finalize_kernel(const float* __restrict__ xs, const float* __restrict__ y,
                        const int* __restrict__ faces,
                        const int* __restrict__ rowmin, const int* __restrict__ colmin,
                        float* __restrict__ out) {
    __shared__ float red[256];
    int b = blockIdx.x;
    int tid = threadIdx.x;

    const float* xb = xs + (size_t)b * V_ * 3;
    const float* yb = y  + (size_t)b * V_ * 3;

    float lsq = 0.0f;
    for (int i = tid; i < V_ * 3; i += 256) {
        float d = xb[i] - yb[i];
        lsq += d * d;
    }

    float volx = 0.0f, voly = 0.0f;
    const int* fb = faces + (size_t)b * F_ * 3;
    for (int f = tid; f < F_; f += 256) {
        int i0 = fb[f * 3 + 0], i1 = fb[f * 3 + 1], i2 = fb[f * 3 + 2];
        {
            const float* a = xb + (size_t)i0 * 3;
            const float* c = xb + (size_t)i1 * 3;
            const float* d = xb + (size_t)i2 * 3;
            float cx = a[1] * c[2] - a[2] * c[1];
            float cy = a[2] * c[0] - a[0] * c[2];
            float cz = a[0] * c[1] - a[1] * c[0];
            volx += (cx * d[0] + cy * d[1] + cz * d[2]) * (1.0f / 6.0f);
        }
        {
            const float* a = yb + (size_t)i0 * 3;
            const float* c = yb + (size_t)i1 * 3;
            const float* d = yb + (size_t)i2 * 3;
            float cx = a[1] * c[2] - a[2] * c[1];
            float cy = a[2] * c[0] - a[0] * c[2];
            float cz = a[0] * c[1] - a[1] * c[0];
            voly += (cx * d[0] + cy * d[1] + cz * d[2]) * (1.0f / 6.0f);
        }
    }

    float chsum = 0.0f;
    const int* rm = rowmin + (size_t)b * P_ * V_;
    for (int i = tid; i < P_ * V_; i += 256) chsum += __int_as_float(rm[i]) * (1.0f / V_);
    const int* cm = colmin + (size_t)b * P_ * E_;
    for (int i = tid; i < P_ * E_; i += 256) chsum += __int_as_float(cm[i]) * (1.0f / E_);

    lsq   = pm_block_reduce(lsq,   red, tid);
    volx  = pm_block_reduce(volx,  red, tid);
    voly  = pm_block_reduce(voly,  red, tid);
    chsum = pm_block_reduce(chsum, red, tid);

    if (tid == 0) {
        float dv = fabsf(voly) - fabsf(volx);
        out[b] = chsum * (1.0f / P_) + lsq + dv * dv;
    }
}

// ---------------------------------------------------------------------------
extern "C" void kernel_launch(void* const* d_in, const int* in_sizes, int n_in,
                              void* d_out, int out_size, void* d_ws, size_t ws_size,
                              hipStream_t stream) {
    const float* xs    = (const float*)d_in[0];
    const float* y     = (const float*)d_in[1];
    const float* pmat  = (const float*)d_in[2];
    const float* em    = (const float*)d_in[3];
    const int*   faces = (const int*)d_in[4];
    float* out = (float*)d_out;

    float* apack  = (float*)d_ws;                          // B*P*V*4 = 524288 f
    float* bpack  = apack  + (size_t)B_ * P_ * V_ * 4;     // B*P*E*4 = 131072 f
    int*   rowmin = (int*)(bpack + (size_t)B_ * P_ * E_ * 4);  // B*P*V = 131072
    int*   colmin = rowmin + (size_t)B_ * P_ * V_;             // B*P*E =  32768

    int prepThreads = B_ * P_ * V_;                        // covers bpack too
    pm_prep_kernel<<<(prepThreads + 255) / 256, 256, 0, stream>>>(y, pmat, em, apack, bpack);
    pm_init_kernel<<<(B_ * P_ * E_ + 255) / 256, 256, 0, stream>>>(colmin);
    pm_chamfer_kernel<<<B_ * P_ * (V_ / ROWS_PER_WG), 256, 0, stream>>>(apack, bpack, rowmin, colmin);
    pm_finalize_kernel<<<B_, 256, 0, stream>>>(xs, y, faces, rowmin, colmin, out);
}